// Lowpass_68418829025363
// MI455X (gfx1250) — compile-verified
//
#include <hip/hip_runtime.h>
#include <cmath>

#ifndef M_PI
#define M_PI 3.14159265358979323846
#endif

typedef __attribute__((ext_vector_type(2)))  float        v2f;
typedef __attribute__((ext_vector_type(4)))  float        v4f;
typedef __attribute__((ext_vector_type(8)))  float        v8f;
typedef __attribute__((ext_vector_type(4)))  unsigned int v4u;
typedef __attribute__((ext_vector_type(4)))  int          v4i;
typedef __attribute__((ext_vector_type(8)))  int          v8i;

#define T_LEN  65536     // samples per channel
#define XOFF   64        // x-buffer starts 64 floats into dynamic LDS (g-table first)
#define NTAPS  32        // pole radius 0.414 -> h[31] ~ 1.6e-12, exact in f32

struct FirParams { float g[NTAPS]; };

// One 256-output window: D(16x16) = T0*X0 + T1*X1 as two independent 4-deep
// WMMA chains (K accumulated 4 at a time). GUARD=true only for window n0==0,
// where the s=1 shifted block reaches before the channel start (zero history).
template <bool GUARD>
__device__ __forceinline__ void fir_window(
    const float* __restrict__ xbuf, float* __restrict__ yp,
    const v2f* __restrict__ A, int n0, int row, int kb)
{
  v8f acc0 = {0.f, 0.f, 0.f, 0.f, 0.f, 0.f, 0.f, 0.f};
  v8f acc1 = {0.f, 0.f, 0.f, 0.f, 0.f, 0.f, 0.f, 0.f};

  #pragma unroll
  for (int t = 0; t < 4; ++t) {
    // s = 0 block: always in range (xo0 >= 0 even at n0 == 0)
    const int xo0 = n0 + (row << 4) + 4 * t + kb;
    v2f b0 = *(const v2f*)(xbuf + xo0);
    acc0 = __builtin_amdgcn_wmma_f32_16x16x4_f32(
        false, A[t], false, b0, (short)0, acc0, false, false);

    // s = 1 block: shifted 16 back; negative only when GUARD && row == 0
    v2f b1 = (v2f){0.f, 0.f};
    if (!GUARD || row >= 1) b1 = *(const v2f*)(xbuf + xo0 - 16);
    acc1 = __builtin_amdgcn_wmma_f32_16x16x4_f32(
        false, A[4 + t], false, b1, (short)0, acc1, false, false);
  }

  const v8f acc = acc0 + acc1;
  __builtin_nontemporal_store((v4f){acc[0], acc[1], acc[2], acc[3]}, (v4f*)yp);
  __builtin_nontemporal_store((v4f){acc[4], acc[5], acc[6], acc[7]}, (v4f*)(yp + 4));
}

// One workgroup per channel. TDM stages the whole 256KB channel into LDS,
// then 8 wave32s compute the 32-tap FIR as blocked-Toeplitz f32 WMMAs.
__global__ __launch_bounds__(256) void lowpass_fir_wmma(
    const float* __restrict__ x, float* __restrict__ y, FirParams P)
{
  extern __shared__ float lds[];
  float* gtab = lds;           // 32 floats of FIR taps
  float* xbuf = lds + XOFF;    // 65536 floats of input

  const int tid = threadIdx.x;
  const long long chBase = (long long)blockIdx.x * T_LEN;

  if (tid < NTAPS) gtab[tid] = P.g[tid];

  // ---- TDM: DMA channel tile (4096 x 16 f32, contiguous) global -> LDS.
  // Scalar branch so only wave 0 issues the descriptor.
  if (__builtin_amdgcn_readfirstlane((int)tid) == 0) {
    unsigned long long ga = (unsigned long long)(const void*)(x + chBase);
    unsigned int ldsAddr  = (unsigned int)(unsigned long long)(void*)xbuf;

    v4u g0;
    g0[0] = 1u;                                      // count=1 valid descriptor
    g0[1] = ldsAddr;                                 // lds_addr (bytes)
    g0[2] = (unsigned int)ga;                        // global_addr[31:0]
    g0[3] = (unsigned int)((ga >> 32) & 0x01FFFFFFu) // global_addr[56:32]
            | (2u << 30);                            // type=2 ("image")

    v8i g1;
    g1[0] = (int)(2u << 16);       // workgroup_mask=0, data_size=2 (4B)
    g1[1] = (int)(4096u << 16);    // tensor_dim0 lo16 = 4096
    g1[2] = (int)(16u << 16);      // tensor_dim0 hi16=0 | tensor_dim1 lo16 = 16
    g1[3] = (int)(4096u << 16);    // tensor_dim1 hi16=0 | tile_dim0 = 4096
    g1[4] = 16;                    // tile_dim1 = 16, tile_dim2 = 0
    g1[5] = 4096;                  // tensor_dim0_stride lo32 = 4096 (contiguous)
    g1[6] = 0;                     // stride0 hi | stride1 lo
    g1[7] = 0;                     // stride1 hi

    v4i gz = {0, 0, 0, 0};
#if defined(__clang_major__) && __clang_major__ >= 23
    v8i gz8 = {0, 0, 0, 0, 0, 0, 0, 0};
    __builtin_amdgcn_tensor_load_to_lds(g0, g1, gz, gz, gz8, 0);
#else
    __builtin_amdgcn_tensor_load_to_lds(g0, g1, gz, gz, 0);
#endif
    __builtin_amdgcn_s_wait_tensorcnt(0);
  }
  __syncthreads();

  // Lane decomposition per ISA f32 WMMA layouts (wave32):
  //   A 16x4:  VGPR0 = K {0 | 2}, VGPR1 = K {1 | 3}  (lanes 0-15 | 16-31), M = lane&15
  //   B 4x16:  same K split, N = lane&15
  //   C/D 16x16: VGPR r -> M = r (+8 for hi lanes), N = lane&15
  const int lane = tid & 31;
  const int hi   = lane >> 4;      // lane-half select
  const int row  = lane & 15;      // A row i / B column j
  const int kb   = hi << 1;        // kappa base: 0 or 2

  // Constant Toeplitz A-operands: T_s[i][m] = g[i - m + 16s], m = 4t + kappa.
  v2f A[8];
  #pragma unroll
  for (int s = 0; s < 2; ++s) {
    #pragma unroll
    for (int t = 0; t < 4; ++t) {
      int i0 = row - (4 * t + kb) + 16 * s;   // kappa = kb     (reg .x)
      int i1 = i0 - 1;                        // kappa = kb + 1 (reg .y)
      float a0 = (i0 >= 0 && i0 < NTAPS) ? gtab[i0] : 0.0f;
      float a1 = (i1 >= 0 && i1 < NTAPS) ? gtab[i1] : 0.0f;
      A[s * 4 + t] = (v2f){a0, a1};
    }
  }

  const int wv     = tid >> 5;                    // wave id 0..7
  const int ybias  = (row << 4) + (hi << 3);      // lane's base inside a window
  float* ychan = y + chBase + ybias;

  // Window 0 (zero history) belongs to wave 0 only: guarded version.
  if (wv == 0) fir_window<true>(xbuf, ychan, A, 0, row, kb);

  // Steady state: all tap indices in range, no guards in the hot loop.
  for (int w = (wv == 0) ? 8 : wv; w < T_LEN / 256; w += 8) {
    const int n0 = w << 8;
    fir_window<false>(xbuf, ychan + n0, A, n0, row, kb);
  }
}

// Host: exact biquad impulse response in double (general form; no a1~0 assumption).
static void compute_fir_taps(float* gOut) {
  const double cutoff = 4000.0, sr = 16000.0, q = 0.707;
  const double w0 = 2.0 * M_PI * cutoff / sr;
  const double alpha = sin(w0) / (2.0 * q);
  const double cw = cos(w0);
  double b0 = (1.0 - cw) / 2.0, b1 = 1.0 - cw, b2 = b0;
  const double a0 = 1.0 + alpha;
  double a1 = -2.0 * cw, a2 = 1.0 - alpha;
  b0 /= a0; b1 /= a0; b2 /= a0; a1 /= a0; a2 /= a0;

  double g[NTAPS];
  for (int n = 0; n < NTAPS; ++n) {
    double v = 0.0;
    if (n == 0) v += b0;
    if (n == 1) v += b1;
    if (n == 2) v += b2;
    if (n >= 1) v -= a1 * g[n - 1];
    if (n >= 2) v -= a2 * g[n - 2];
    g[n] = v;
  }
  for (int n = 0; n < NTAPS; ++n) gOut[n] = (float)g[n];
}

extern "C" void kernel_launch(void* const* d_in, const int* in_sizes, int n_in,
                              void* d_out, int out_size, void* d_ws, size_t ws_size,
                              hipStream_t stream) {
  const float* x = (const float*)d_in[0];
  float* y = (float*)d_out;
  const int nch = in_sizes[0] / T_LEN;   // 32 * 16 = 512 channels

  FirParams P;
  compute_fir_taps(P.g);

  const size_t smem = (size_t)(XOFF + T_LEN) * sizeof(float);  // 262400 B
  (void)hipFuncSetAttribute((const void*)lowpass_fir_wmma,
                            hipFuncAttributeMaxDynamicSharedMemorySize,
                            (int)smem);

  lowpass_fir_wmma<<<dim3(nch), dim3(256), smem, stream>>>(x, y, P);

  (void)n_in; (void)out_size; (void)d_ws; (void)ws_size;
}